// MaxWeightGNN_23476291240206
// MI455X (gfx1250) — compile-verified
//
#include <hip/hip_runtime.h>

// ---------------------------------------------------------------------------
// MaxWeightGNN on MI455X (gfx1250):
//   xp = prod(x, axis=1); scatter-max over edges; self-loop max; [N,2]@[2,1].
// Memory-bound: 256MB edge stream dominates (~11us @ 23.3TB/s). xp/key (1MB
// each) live in L2 (192MB), so gathers + atomic-max resolve in-cache.
// CDNA5 paths used: async global->LDS edge streaming (ASYNCcnt), u32 atomic
// max with monotone float keys, and a v_wmma_f32_16x16x4_f32 final combine.
// ---------------------------------------------------------------------------

typedef __attribute__((ext_vector_type(2))) float v2f;
typedef __attribute__((ext_vector_type(4))) float v4f;
typedef __attribute__((ext_vector_type(8))) float v8f;
typedef int vint4 __attribute__((vector_size(16)));  // matches builtin param

#define AS1 __attribute__((address_space(1)))
#define AS3 __attribute__((address_space(3)))

// Monotone float -> uint mapping: a <= b (as floats)  <=>  fkey(a) <= fkey(b).
__device__ __forceinline__ unsigned fkey(float f) {
  unsigned b = __float_as_uint(f);
  return (b & 0x80000000u) ? ~b : (b | 0x80000000u);
}
__device__ __forceinline__ float funkey(unsigned k) {
  unsigned b = (k & 0x80000000u) ? (k & 0x7FFFFFFFu) : ~k;
  return __uint_as_float(b);
}
#define KEY_NEG_INF 0x007FFFFFu  // fkey(-inf): segment_max identity

// ---------------------------------------------------------------------------
// Kernel 1: xp[i] = x[i,0]*x[i,1];  key[i] = fkey(-inf)
// ---------------------------------------------------------------------------
__global__ __launch_bounds__(256)
void gnn_init_kernel(const float2* __restrict__ x, float* __restrict__ xp,
                     unsigned* __restrict__ key, int n) {
  int i = blockIdx.x * blockDim.x + threadIdx.x;
  if (i < n) {
    float2 v = x[i];
    xp[i] = v.x * v.y;
    key[i] = KEY_NEG_INF;
  }
}

// ---------------------------------------------------------------------------
// Kernel 2: scatter-max over edges. Double-buffered async global->LDS stream
// of src/dst int64 indices (16B = 2 edges per lane per async b128).
// ---------------------------------------------------------------------------
#define EB_THREADS 256
#define CHUNK 512  // edges per chunk (2 per thread); 4KB per index array

__device__ __forceinline__ void async_copy16(const void* g, void* lds) {
  __builtin_amdgcn_global_load_async_to_lds_b128(
      (AS1 vint4*)g, (AS3 vint4*)lds, 0, 0);
}

__global__ __launch_bounds__(EB_THREADS)
void gnn_edge_kernel(const unsigned long long* __restrict__ src,
                     const unsigned long long* __restrict__ dst,
                     const float* __restrict__ xp,
                     unsigned* __restrict__ key,
                     long long edges_per_block) {
  __shared__ unsigned long long s_src[2][CHUNK];
  __shared__ unsigned long long s_dst[2][CHUNK];

  const int t = threadIdx.x;
  const long long base = (long long)blockIdx.x * edges_per_block;
  const int niter = (int)(edges_per_block / CHUNK);

  // Prefetch chunk 0 into buffer 0 (2 async ops per lane).
  async_copy16(src + base + 2 * t, &s_src[0][2 * t]);
  async_copy16(dst + base + 2 * t, &s_dst[0][2 * t]);

  for (int it = 0; it < niter; ++it) {
    const int buf = it & 1;
    if (it + 1 < niter) {
      // Prefetch next chunk into the other buffer, then wait for current
      // chunk's 2 copies (async loads complete in order per wave).
      const long long eb = base + (long long)(it + 1) * CHUNK;
      async_copy16(src + eb + 2 * t, &s_src[buf ^ 1][2 * t]);
      async_copy16(dst + eb + 2 * t, &s_dst[buf ^ 1][2 * t]);
      __builtin_amdgcn_s_wait_asynccnt(2);
    } else {
      __builtin_amdgcn_s_wait_asynccnt(0);
    }
    __syncthreads();  // all waves' copies for chunk `it` now visible in LDS

#pragma unroll
    for (int k = 0; k < 2; ++k) {
      unsigned s = (unsigned)s_src[buf][2 * t + k];  // node ids < 2^18
      unsigned d = (unsigned)s_dst[buf][2 * t + k];
      atomicMax(&key[d], fkey(xp[s]));  // global_atomic_max_u32, hits L2
    }
    __syncthreads();  // protect buf before it is overwritten next iter
  }
}

// ---------------------------------------------------------------------------
// Kernel 3: z = [xp, max(neigh_max, xp)] @ [w0, w1]^T via WMMA 16x16x4 f32.
// Per wave: 16 nodes. A (16x4): lanes 0-15 hold row M=lane, VGPR0=K0 (xp),
// VGPR1=K1 (agg); lanes 16-31 hold K2,K3 = 0. B (4x16): VGPR0 lanes0-15 =
// row K0 (w0), lanes16-31 = K2 (0); VGPR1 lanes0-15 = K1 (w1). D[m][n] =
// xp[m]*w0 + agg[m]*w1 for all n; read column 0 from lanes 0 and 16.
// ---------------------------------------------------------------------------
__global__ __launch_bounds__(256)
void gnn_final_kernel(const float* __restrict__ xp,
                      const unsigned* __restrict__ key,
                      const float* __restrict__ w,
                      float* __restrict__ out, int n) {
  const int lane = threadIdx.x & 31;
  const int wave = (blockIdx.x * blockDim.x + threadIdx.x) >> 5;
  const int base = wave * 16;
  const float w0 = w[0];
  const float w1 = w[1];

  int idx = base + (lane & 15);
  idx = idx < n ? idx : n - 1;  // clamp (n is a multiple of 16 here)
  float p = xp[idx];
  float m = fmaxf(funkey(key[idx]), p);  // self-loop max; -inf decodes exactly

  const bool lo = (lane < 16);
  v2f A;
  A.x = lo ? p : 0.0f;
  A.y = lo ? m : 0.0f;
  v2f B;
  B.x = lo ? w0 : 0.0f;
  B.y = lo ? w1 : 0.0f;
  v8f C = {};

  // EXEC is all-ones here (no divergent branch encloses this point).
  v8f D = __builtin_amdgcn_wmma_f32_16x16x4_f32(
      /*neg_a=*/false, A, /*neg_b=*/false, B,
      /*c_mod=*/(short)0, C, /*reuse_a=*/false, /*reuse_b=*/false);

  // D VGPR r: lanes 0-15 -> M=r (N=lane), lanes 16-31 -> M=r+8.
  // Lane 0 holds z[base..base+7]; lane 16 holds z[base+8..base+15].
  // Store as two coalesced b128s under one guard (n % 16 == 0 here).
  const int ob = base + (lane == 16 ? 8 : 0);
  if ((lane == 0 || lane == 16) && (ob + 8 <= n)) {
    v4f lo4, hi4;
    lo4.x = D[0]; lo4.y = D[1]; lo4.z = D[2]; lo4.w = D[3];
    hi4.x = D[4]; hi4.y = D[5]; hi4.z = D[6]; hi4.w = D[7];
    v4f* o = (v4f*)(out + ob);
    o[0] = lo4;
    o[1] = hi4;
  }
}

// ---------------------------------------------------------------------------
// Launch
// ---------------------------------------------------------------------------
extern "C" void kernel_launch(void* const* d_in, const int* in_sizes, int n_in,
                              void* d_out, int out_size, void* d_ws,
                              size_t ws_size, hipStream_t stream) {
  const float* x = (const float*)d_in[0];                       // [N,2] f32
  const unsigned long long* ei = (const unsigned long long*)d_in[1];  // [2,E] i64
  const float* w = (const float*)d_in[2];                       // [1,2] f32
  float* out = (float*)d_out;                                   // [N,1] f32

  const int n = in_sizes[0] / 2;            // 262144
  const long long e = in_sizes[1] / 2;      // 16777216

  float* xp = (float*)d_ws;                                   // n floats
  unsigned* key = (unsigned*)((char*)d_ws + (size_t)n * sizeof(float));

  // 1) products + key init
  gnn_init_kernel<<<(n + 255) / 256, 256, 0, stream>>>((const float2*)x, xp,
                                                       key, n);

  // 2) scatter-max over edges: 1024 blocks x 16384 contiguous edges each,
  //    32 double-buffered async chunks of 512 edges.
  const int eblocks = 1024;
  const long long epb = e / eblocks;  // 16384, multiple of CHUNK
  gnn_edge_kernel<<<eblocks, EB_THREADS, 0, stream>>>(ei, ei + e, xp, key,
                                                      epb);

  // 3) finalize: one WMMA per wave, 16 nodes per wave.
  const int waves = (n + 15) / 16;     // 16384
  const int threads = waves * 32;      // 524288
  gnn_final_kernel<<<(threads + 255) / 256, 256, 0, stream>>>(xp, key, w, out,
                                                              n);
}